// percnn_convective_flow_71133248357054
// MI455X (gfx1250) — compile-verified
//
#include <hip/hip_runtime.h>

// ---------------------------------------------------------------------------
// percnn convective flow, collapsed form.
//   grads[k](s) = A_k s^2 + B_k s + C_k   (s = u for k=0,1 ; s = v for k=2,3)
//   out_u = u*g0(u) + v*g1(u)
//   out_v = u*g2(v) + v*g3(v)
// Memory-bound: 32 MiB traffic -> ~1.4 us floor @ 23.3 TB/s.
// Uses gfx1250 async global->LDS staging (ASYNCcnt) for the streaming loads.
// ---------------------------------------------------------------------------

#define HWF  262144u   // 512*512 floats per plane
#define HW4  65536u    // float4 per plane
#define NB4  524288u   // 8 batches * HW4  (== 2048 blocks * 256 threads)

struct alignas(16) f4 { float x, y, z, w; };

// --- Pass 1: reduce weights to 12 polynomial coefficients ------------------
__global__ void percnn_coeff_kernel(const float* __restrict__ W1,
                                    const float* __restrict__ B1,
                                    const float* __restrict__ W2,
                                    const float* __restrict__ B2,
                                    const float* __restrict__ W3,
                                    const float* __restrict__ B3,
                                    float* __restrict__ cf) {
    const int k = threadIdx.x;
    if (k < 4) {
        float A = 0.f, Bq = 0.f, C = 0.f;
        #pragma unroll
        for (int c = 0; c < 16; ++c) {
            const float w1 = W1[k * 16 + c], b1 = B1[k * 16 + c];
            const float w2 = W2[k * 16 + c], b2 = B2[k * 16 + c];
            const float w3 = W3[k * 16 + c];
            A  = fmaf(w1 * w2, w3, A);
            Bq = fmaf(fmaf(w1, b2, w2 * b1), w3, Bq);
            C  = fmaf(b1 * b2, w3, C);
        }
        cf[k]     = A;
        cf[4 + k] = Bq;
        cf[8 + k] = C + B3[k];
    }
}

// --- Pass 2: streaming elementwise kernel ----------------------------------
__global__ void __launch_bounds__(256)
percnn_main_kernel(const float* __restrict__ x,
                   const float* __restrict__ cf,
                   float* __restrict__ out) {
    __shared__ f4 su[256];
    __shared__ f4 sv[256];

    const unsigned tid = threadIdx.x;
    const unsigned i   = blockIdx.x * 256u + tid;     // exact cover, no tail
    const unsigned b   = i >> 16;                      // batch
    const unsigned j   = i & 65535u;                   // float4 index in plane
    const unsigned long long base =
        (unsigned long long)b * (2ull * HWF) + (unsigned long long)j * 4ull;

    const float* up = x + base;          // u plane
    const float* vp = up + HWF;          // v plane

    // LDS byte offsets for this lane's private slots (flat addr low 32 bits).
    const unsigned lds_u = (unsigned)(unsigned long long)&su[tid];
    const unsigned lds_v = (unsigned)(unsigned long long)&sv[tid];

    // Async global -> LDS staging (tracked by ASYNCcnt, no VGPR dest).
    asm volatile(
        "global_load_async_to_lds_b128 %0, %1, off\n\t"
        "global_load_async_to_lds_b128 %2, %3, off"
        :
        : "v"(lds_u), "v"(up), "v"(lds_v), "v"(vp)
        : "memory");

    // Uniform coefficient fetch overlaps the async copies (scalar loads).
    const float A0 = cf[0], A1 = cf[1], A2 = cf[2], A3 = cf[3];
    const float Q0 = cf[4], Q1 = cf[5], Q2 = cf[6], Q3 = cf[7];
    const float C0 = cf[8], C1 = cf[9], C2 = cf[10], C3 = cf[11];

    asm volatile("s_wait_asynccnt 0" ::: "memory");

    const f4 u = su[tid];   // ds_load_b128
    const f4 v = sv[tid];

    f4 ou, ov;
    {
        const float* uu = &u.x;
        const float* vv = &v.x;
        float* pu = &ou.x;
        float* pv = &ov.x;
        #pragma unroll
        for (int e = 0; e < 4; ++e) {
            const float ue = uu[e], ve = vv[e];
            const float g0 = fmaf(fmaf(A0, ue, Q0), ue, C0);
            const float g1 = fmaf(fmaf(A1, ue, Q1), ue, C1);
            const float g2 = fmaf(fmaf(A2, ve, Q2), ve, C2);
            const float g3 = fmaf(fmaf(A3, ve, Q3), ve, C3);
            pu[e] = fmaf(ue, g0, ve * g1);
            pv[e] = fmaf(ue, g2, ve * g3);
        }
    }

    *(f4*)(out + base)       = ou;   // global_store_b128
    *(f4*)(out + base + HWF) = ov;
}

// ---------------------------------------------------------------------------
extern "C" void kernel_launch(void* const* d_in, const int* in_sizes, int n_in,
                              void* d_out, int out_size, void* d_ws, size_t ws_size,
                              hipStream_t stream) {
    const float* x  = (const float*)d_in[0];
    const float* W1 = (const float*)d_in[1];
    const float* B1 = (const float*)d_in[2];
    const float* W2 = (const float*)d_in[3];
    const float* B2 = (const float*)d_in[4];
    const float* W3 = (const float*)d_in[5];
    const float* B3 = (const float*)d_in[6];
    float* cf  = (float*)d_ws;          // 12 floats of scratch
    float* out = (float*)d_out;

    percnn_coeff_kernel<<<1, 32, 0, stream>>>(W1, B1, W2, B2, W3, B3, cf);
    percnn_main_kernel<<<NB4 / 256u, 256, 0, stream>>>(x, cf, out);
}